// ObjectCaps_32641751449676
// MI455X (gfx1250) — compile-verified
//
#include <hip/hip_runtime.h>

typedef __attribute__((ext_vector_type(2))) float v2f;
typedef __attribute__((ext_vector_type(4))) float v4f;
typedef __attribute__((ext_vector_type(8))) float v8f;

#define B_TOTAL 512
#define R_TOTAL 1152
#define CIN     8
#define J_CAPS  10
#define O_DIM   16
#define ITERS   3

#define NB      16                  // batches per block == WMMA N
#define WAVES   8                   // waves per block (wave32)
#define THREADS (WAVES * 32)
#define RBLK    16                  // route-blocks per batch-tile
#define RPB     (R_TOTAL / RBLK)    // 72 routes per block -> 9 per wave

// sched_barrier mask: allow SALU (0x4) + all VMEM (0x10|0x20|0x40) to cross,
// block VALU/WMMA reordering -> caps transient register pressure < 256 VGPRs
// so the compiler avoids s_set_vgpr_msb bank switching in the hot loop.
#define SCHED_FENCE() __builtin_amdgcn_sched_barrier(0x0074)

// ---------------------------------------------------------------------------
// Zero the workspace (s accumulator + vsum), contiguous.
// ---------------------------------------------------------------------------
__global__ void caps_zero_kernel(float* __restrict__ p, int n) {
    int i = blockIdx.x * blockDim.x + threadIdx.x;
    if (i < n) p[i] = 0.0f;
}

// ---------------------------------------------------------------------------
// One routing pass. u_hat is recomputed on the matrix pipe (never stored):
//   pass 1: u = W[r] @ x  (V_WMMA_F32_16X16X4_F32), logit_j = u . vsum
//           (manual depth-2 pipeline: WMMA for j issues before the dot of j-1)
//   softmax over j
//   pass 2: s_tile[j] = WMMA(W_j[r], c_j * x, s_tile[j])
//           (c_j folded into the B operand; B lane index == batch index)
//
// K-chunk remap: half-0 lanes own K={0..3}, half-1 lanes K={4..7}, so each
// lane's two v2f WMMA operands are one contiguous, 16B-aligned b128 load.
//
// grid = (B/NB, RBLK), block = 256 (8 wave32)
// ---------------------------------------------------------------------------
__global__ __launch_bounds__(THREADS)
void caps_iter_kernel(const float* __restrict__ x,     // [B, R, CIN]
                      const float* __restrict__ W,     // [R, J, O, CIN]
                      const float* __restrict__ vsum,  // [B, J, O]  (sum of v_t so far)
                      float* __restrict__ s_out,       // [B, J, O]  (zeroed)
                      int first)                       // it==0: softmax is uniform
{
    const int lane   = threadIdx.x & 31;
    const int wave   = threadIdx.x >> 5;
    const int half   = lane >> 4;    // lane half: selects o-range and K-range
    const int lane16 = lane & 15;    // WMMA M/N row index
    const int b      = blockIdx.x * NB + lane16;
    const int r_beg  = blockIdx.y * RPB + wave;
    const int r_end  = blockIdx.y * RPB + RPB;

    // Per-lane cache of vsum: this lane's batch, o = 8*half + 0..7.
    float vreg[J_CAPS][8];
    if (!first) {
        const float* vp = vsum + (size_t)b * (J_CAPS * O_DIM) + 8 * half;
        #pragma unroll
        for (int j = 0; j < J_CAPS; ++j)
            #pragma unroll
            for (int p = 0; p < 8; ++p)
                vreg[j][p] = vp[j * O_DIM + p];
    }

    // Persistent s accumulators ARE the WMMA C/D tiles:
    // s_tile[j][p] = s[b = b0+lane16, j, o = p + 8*half]
    v8f s_tile[J_CAPS];
    #pragma unroll
    for (int j = 0; j < J_CAPS; ++j)
        s_tile[j] = (v8f){0.f, 0.f, 0.f, 0.f, 0.f, 0.f, 0.f, 0.f};

    for (int r = r_beg; r < r_end; r += WAVES) {
        // ---- B operand: x[b, r, 4*half .. 4*half+3] : one b128 per lane.
        const float* xr = x + ((size_t)b * R_TOTAL + r) * CIN;
        const v4f bm  = *(const v4f*)(xr + 4 * half);
        const v2f bm0 = {bm.x, bm.y};   // K-chunk 0 (this lane's first 2 K)
        const v2f bm1 = {bm.z, bm.w};   // K-chunk 1

        const float* Wr = W + (size_t)r * (J_CAPS * O_DIM * CIN);
        __builtin_prefetch(Wr + WAVES * (J_CAPS * O_DIM * CIN), 0, 1);

        float cw[J_CAPS];
        if (!first) {
            // ---- pass 1: logits b_ij = u_hat . vsum, depth-2 pipelined
            float mx = -1e30f;
            v8f accP;
            {   // prologue: WMMA pair for j = 0
                const float* Wp = Wr + (size_t)lane16 * CIN + 4 * half;
                const v4f a  = *(const v4f*)Wp;
                const v2f a0 = {a.x, a.y};
                const v2f a1 = {a.z, a.w};
                v8f t = {0.f, 0.f, 0.f, 0.f, 0.f, 0.f, 0.f, 0.f};
                t = __builtin_amdgcn_wmma_f32_16x16x4_f32(
                        false, a0, false, bm0, (short)0, t, false, false);
                accP = __builtin_amdgcn_wmma_f32_16x16x4_f32(
                        false, a1, false, bm1, (short)0, t, false, false);
            }
            #pragma unroll
            for (int j = 1; j < J_CAPS; ++j) {
                // issue WMMA pair for j ...
                const float* Wp = Wr + (size_t)(j * O_DIM + lane16) * CIN + 4 * half;
                const v4f a  = *(const v4f*)Wp;
                const v2f a0 = {a.x, a.y};
                const v2f a1 = {a.z, a.w};
                v8f t = {0.f, 0.f, 0.f, 0.f, 0.f, 0.f, 0.f, 0.f};
                t = __builtin_amdgcn_wmma_f32_16x16x4_f32(
                        false, a0, false, bm0, (short)0, t, false, false);
                t = __builtin_amdgcn_wmma_f32_16x16x4_f32(
                        false, a1, false, bm1, (short)0, t, false, false);
                // ... then consume tile j-1 (its latency hidden by pair j)
                float part = 0.0f;
                #pragma unroll
                for (int p = 0; p < 8; ++p)
                    part = fmaf(accP[p], vreg[j - 1][p], part);
                part += __shfl_xor(part, 16, 32);    // combine the two o-halves
                cw[j - 1] = part;
                mx = fmaxf(mx, part);
                accP = t;
                SCHED_FENCE();   // cap live accumulator tiles at 2
            }
            {   // epilogue: consume tile j = 9
                float part = 0.0f;
                #pragma unroll
                for (int p = 0; p < 8; ++p)
                    part = fmaf(accP[p], vreg[J_CAPS - 1][p], part);
                part += __shfl_xor(part, 16, 32);
                cw[J_CAPS - 1] = part;
                mx = fmaxf(mx, part);
            }
            // softmax over capsules j
            float ssum = 0.0f;
            #pragma unroll
            for (int j = 0; j < J_CAPS; ++j) { cw[j] = __expf(cw[j] - mx); ssum += cw[j]; }
            const float inv = 1.0f / ssum;
            #pragma unroll
            for (int j = 0; j < J_CAPS; ++j) cw[j] *= inv;
        } else {
            // b_ij == 0 -> softmax is exactly uniform (matches reference it 0)
            #pragma unroll
            for (int j = 0; j < J_CAPS; ++j) cw[j] = 1.0f / J_CAPS;
        }

        // ---- pass 2: s_tile[j] += W_j[r] @ (c_j * x) on the matrix pipe
        #pragma unroll
        for (int j = 0; j < J_CAPS; ++j) {
            const float* Wp = Wr + (size_t)(j * O_DIM + lane16) * CIN + 4 * half;
            const v4f a  = *(const v4f*)Wp;              // CSE/WGP$ hit from pass 1
            const v2f a0 = {a.x, a.y};
            const v2f a1 = {a.z, a.w};
            const float c = cw[j];
            const v2f bj0 = {bm.x * c, bm.y * c};        // scale B by c_j (pk_mul)
            const v2f bj1 = {bm.z * c, bm.w * c};
            s_tile[j] = __builtin_amdgcn_wmma_f32_16x16x4_f32(
                            false, a0, false, bj0, (short)0, s_tile[j], false, false);
            s_tile[j] = __builtin_amdgcn_wmma_f32_16x16x4_f32(
                            false, a1, false, bj1, (short)0, s_tile[j], false, false);
            SCHED_FENCE();   // no consumer stall here; just caps B-op transients
        }
    }

    // ---- cross-block reduction over route-chunks (global_atomic_add_f32)
    float* sp = s_out + (size_t)b * (J_CAPS * O_DIM) + 8 * half;
    #pragma unroll
    for (int j = 0; j < J_CAPS; ++j)
        #pragma unroll
        for (int p = 0; p < 8; ++p)
            atomicAdd(sp + j * O_DIM + p, s_tile[j][p]);
}

// ---------------------------------------------------------------------------
// Squash: v = (|s|^2/(1+|s|^2)) * s * rsqrt(|s|^2 + 1e-8), one thread per (b,j).
// Non-final iters: vsum += v and re-zero s for the next pass.
// Final iter: write v to d_out ([B,J,O,1] == flat [B,J,O]).
// ---------------------------------------------------------------------------
__global__ void caps_squash_kernel(float* __restrict__ s,
                                   float* __restrict__ vsum,
                                   float* __restrict__ out,
                                   int last)
{
    int idx = blockIdx.x * blockDim.x + threadIdx.x;   // idx = b*J + j
    if (idx >= B_TOTAL * J_CAPS) return;

    float* sp = s + (size_t)idx * O_DIM;
    float sv[O_DIM];
    float sq = 0.0f;
    #pragma unroll
    for (int o = 0; o < O_DIM; ++o) { sv[o] = sp[o]; sq = fmaf(sv[o], sv[o], sq); }
    const float coef = (sq / (1.0f + sq)) * rsqrtf(sq + 1e-8f);

    if (last) {
        float* op = out + (size_t)idx * O_DIM;
        #pragma unroll
        for (int o = 0; o < O_DIM; ++o) op[o] = coef * sv[o];
    } else {
        float* vp = vsum + (size_t)idx * O_DIM;
        #pragma unroll
        for (int o = 0; o < O_DIM; ++o) {
            vp[o] += coef * sv[o];   // accumulate v_t (b_ij is linear in sum of v_t)
            sp[o]  = 0.0f;           // re-zero s for next routing pass
        }
    }
}

// ---------------------------------------------------------------------------
extern "C" void kernel_launch(void* const* d_in, const int* in_sizes, int n_in,
                              void* d_out, int out_size, void* d_ws, size_t ws_size,
                              hipStream_t stream) {
    const float* x = (const float*)d_in[0];   // [512, 1152, 8]
    const float* W = (const float*)d_in[1];   // [1152, 10, 16, 8]
    float* out  = (float*)d_out;              // [512, 10, 16, 1]
    float* s    = (float*)d_ws;                         // [B,J,O] partial sums
    float* vsum = s + (size_t)B_TOTAL * J_CAPS * O_DIM; // [B,J,O] running sum of v_t

    const int nz = 2 * B_TOTAL * J_CAPS * O_DIM;        // zero s and vsum
    caps_zero_kernel<<<(nz + 255) / 256, 256, 0, stream>>>(s, nz);

    for (int it = 0; it < ITERS; ++it) {
        caps_iter_kernel<<<dim3(B_TOTAL / NB, RBLK), THREADS, 0, stream>>>(
            x, W, vsum, s, (it == 0) ? 1 : 0);
        caps_squash_kernel<<<(B_TOTAL * J_CAPS + 255) / 256, 256, 0, stream>>>(
            s, vsum, out, (it == ITERS - 1) ? 1 : 0);
    }
}